// Cerealbar_VIN_2783138808251
// MI455X (gfx1250) — compile-verified
//
#include <hip/hip_runtime.h>
#include <stdint.h>

// Hex-grid VIN. One workgroup per batch element; entire VI state lives in LDS.
// CDNA5-specific paths used: global_load_async_to_lds_b64/b32 (+ s_wait_asynccnt)
// for the per-block input staging.

#define NROT 6
#define EDIM 25
#define HDIM 37            // E + (E-1)/2
#define HP   (HDIM + 2)    // 39  (zero-padded rows)
#define EP   (EDIM + 2)    // 27  (zero-padded cols)
#define SPLANE (HP * EP)          // 1053
#define SSIZE  (NROT * SPLANE)    // 6318 floats per buffer
#define PLANE  (HDIM * EDIM)      // 925 interior cells
#define GOALS_PER_B (NROT * EDIM * EDIM)  // 3750 floats (15000 B, 8B aligned per batch)
#define OBST_PER_B  (EDIM * EDIM)         // 625 floats  (2500 B, only 4B aligned)
#define NTHREADS 256
#define NCHUNK 4            // ceil(925/256)

// Action-0 hex shift per rotation (conv kernel position minus (1,1)).
// Action 1 is the exact negation; actions 2/3 are center taps with rotated channel.
constexpr int kDU0[NROT] = {0, 1, 1, 0, -1, -1};
constexpr int kDC0[NROT] = {1, 0, -1, -1, 0, 1};

extern "C" __global__ __launch_bounds__(NTHREADS)
void vin_iter_kernel(const float* __restrict__ goals,     // (B,6,25,25)
                     const int*   __restrict__ state,     // (B,3)
                     const float* __restrict__ obst,      // (B,25,25)
                     const int*   __restrict__ niter_ptr, // scalar
                     float*       __restrict__ out)       // (B,4)
{
    __shared__ float sA[SSIZE];   // VI buffer A (zero border)
    __shared__ float sB[SSIZE];   // VI buffer B; doubles as input staging at startup
    __shared__ float sM[PLANE];   // M = (1-obstacle)*axial_mask, {0,1}

    const int b   = blockIdx.x;
    const int tid = threadIdx.x;

    // ---------- Phase 1: async global -> LDS staging of this block's inputs ----------
    {
        const unsigned ldsBase = (unsigned)(uintptr_t)(void*)sB;
        // goals: 3750 dwords = 1875 qwords, both global and LDS sides 8B aligned
        const float* gsrc = goals + (size_t)b * GOALS_PER_B;   // uniform -> SGPR pair
        for (int i = tid; i < GOALS_PER_B / 2; i += NTHREADS) {
            unsigned lds_addr = ldsBase + (unsigned)(i * 8);
            unsigned voff     = (unsigned)(i * 8);
            asm volatile("global_load_async_to_lds_b64 %0, %1, %2"
                         :: "v"(lds_addr), "v"(voff), "s"(gsrc) : "memory");
        }
        // obstacles: per-batch base only 4B aligned -> b32
        const float* osrc = obst + (size_t)b * OBST_PER_B;
        const unsigned obase = ldsBase + (unsigned)(GOALS_PER_B * 4);
        for (int i = tid; i < OBST_PER_B; i += NTHREADS) {
            unsigned lds_addr = obase + (unsigned)(i * 4);
            unsigned voff     = (unsigned)(i * 4);
            asm volatile("global_load_async_to_lds_b32 %0, %1, %2"
                         :: "v"(lds_addr), "v"(voff), "s"(osrc) : "memory");
        }
    }

    // ---------- Phase 2: zero buffer A while async copies are in flight ----------
    for (int i = tid; i < SSIZE; i += NTHREADS) sA[i] = 0.0f;

    asm volatile("s_wait_asynccnt 0" ::: "memory");
    __syncthreads();

    // ---------- Phase 3: build M, per-thread gm/0.9*M registers, init sA = gm ----------
    const float* stageG = sB;                 // goals (6,25,25) row-major
    const float* stageO = sB + GOALS_PER_B;   // obstacles (25,25)

    float gmreg[NCHUNK][NROT];   // gm = goals_axial * M  (held in registers)
    float mgreg[NCHUNK];         // 0.9f * M
    int   pcr  [NCHUNK];         // padded-plane offset of the cell

    #pragma unroll
    for (int k = 0; k < NCHUNK; ++k) {
        mgreg[k] = 0.0f;
        pcr[k]   = EP + 1;
        #pragma unroll
        for (int r = 0; r < NROT; ++r) gmreg[k][r] = 0.0f;

        int idx = tid + k * NTHREADS;
        if (idx < PLANE) {
            int u  = idx / EDIM;
            int c  = idx - u * EDIM;
            int pc = (u + 1) * EP + (c + 1);
            pcr[k] = pc;
            int i  = u + (c >> 1) - 12;             // axial -> offset-grid row
            bool mapped = (i >= 0) && (i < EDIM);
            float m = mapped ? (1.0f - stageO[i * EDIM + c]) : 0.0f;
            sM[idx]  = m;
            mgreg[k] = 0.9f * m;
            #pragma unroll
            for (int r = 0; r < NROT; ++r) {
                float g   = mapped ? stageG[r * (EDIM * EDIM) + i * EDIM + c] : 0.0f;
                float val = g * m;                  // gm
                gmreg[k][r] = val;
                sA[r * SPLANE + pc] = val;          // s_0 encodes v = 0 -> s = gm
            }
        }
    }
    __syncthreads();

    // ---------- Phase 4: zero buffer B (staging now consumed) ----------
    for (int i = tid; i < SSIZE; i += NTHREADS) sB[i] = 0.0f;
    __syncthreads();

    // ---------- Phase 5: value-iteration; s' = gm + 0.9*M*max4(shifted s) ----------
    const int niter = niter_ptr[0];
    float* cur = sA;
    float* nxt = sB;
    for (int it = 0; it < niter; ++it) {
        #pragma unroll
        for (int k = 0; k < NCHUNK; ++k) {
            if (tid + k * NTHREADS < PLANE) {
                const int   pc = pcr[k];
                const float mg = mgreg[k];
                // Center taps (actions 2/3) are shared across rotations: load once.
                float cen[NROT];
                #pragma unroll
                for (int r = 0; r < NROT; ++r) cen[r] = cur[r * SPLANE + pc];
                #pragma unroll
                for (int r = 0; r < NROT; ++r) {
                    const int off = kDU0[r] * EP + kDC0[r];
                    const float* pr = cur + r * SPLANE;
                    float m0 = pr[pc + off];                      // action 0 (fwd)
                    float m1 = pr[pc - off];                      // action 1 (bwd)
                    float t  = fmaxf(fmaxf(m0, m1),
                                     fmaxf(cen[(r + 1) % NROT],  // rotate left
                                           cen[(r + 5) % NROT]));// rotate right
                    nxt[r * SPLANE + pc] = __builtin_fmaf(mg, t, gmreg[k][r]);
                }
            }
        }
        __syncthreads();
        float* tswap = cur; cur = nxt; nxt = tswap;
    }

    // ---------- Phase 6: q_niter at the query cell; 4 lanes = 4 actions ----------
    if (tid < 4) {
        int alpha = state[b * 3 + 0];
        int u0    = state[b * 3 + 1];
        int v0    = state[b * 3 + 2];
        int rot   = (alpha + 1) % NROT;
        int uu    = u0 - (v0 >> 1) + 12;
        int pc    = (uu + 1) * EP + (v0 + 1);
        float mcell = sM[uu * EDIM + v0];
        int off   = kDU0[rot] * EP + kDC0[rot];
        float q;
        if      (tid == 0) q = cur[rot * SPLANE + pc + off];
        else if (tid == 1) q = cur[rot * SPLANE + pc - off];
        else if (tid == 2) q = cur[((rot + 1) % NROT) * SPLANE + pc];
        else               q = cur[((rot + 5) % NROT) * SPLANE + pc];
        out[b * 4 + tid] = q * mcell;
    }
}

extern "C" void kernel_launch(void* const* d_in, const int* in_sizes, int n_in,
                              void* d_out, int out_size, void* d_ws, size_t ws_size,
                              hipStream_t stream) {
    const float* goals = (const float*)d_in[0];
    const int*   state = (const int*)  d_in[1];
    const float* obst  = (const float*)d_in[2];
    const int*   nit   = (const int*)  d_in[3];
    const int B = in_sizes[0] / GOALS_PER_B;   // 512
    vin_iter_kernel<<<dim3(B), dim3(NTHREADS), 0, stream>>>(
        goals, state, obst, nit, (float*)d_out);
}